// knn_ContrastiveLoss_1580547971567
// MI455X (gfx1250) — compile-verified
//
#include <hip/hip_runtime.h>

typedef __attribute__((ext_vector_type(2))) float v2f;
typedef __attribute__((ext_vector_type(8))) float v8f;

#define D_DIM 256
#define MARGIN_V 0.2f
#define ROWS_PER_WAVE 16
#define WAVES_PER_BLOCK 8
#define ROWS_PER_BLOCK (ROWS_PER_WAVE * WAVES_PER_BLOCK)  // 128
#define PF 4  // software-pipeline depth, in K=4 steps (12 loads in flight)

// Each wave computes 16 rowwise dot products for both (im,s) and (knn_im,s)
// using V_WMMA_F32_16X16X4_F32 as the reduction engine:
//   acc (16x16) += A(16x4) * B(4x16), iterated K=4 at a time over D=256.
// With A = im-tile and B = s-tile^T, diag(acc) holds the 16 rowwise dots.
// 4-deep register double-buffering keeps 12 b64 loads in flight per wave so
// the legalizer emits partial s_wait_loadcnt instead of draining to 0.
__global__ __launch_bounds__(ROWS_PER_BLOCK / ROWS_PER_WAVE * 32)
void knn_contrastive_wmma(const float* __restrict__ im,
                          const float* __restrict__ knn_im,
                          const float* __restrict__ s,
                          float* __restrict__ out,
                          int n_rows) {
  const int lane = (int)(threadIdx.x & 31u);
  const int wave = (int)(threadIdx.x >> 5);
  const long long baseRow = (long long)blockIdx.x * ROWS_PER_BLOCK +
                            (long long)wave * ROWS_PER_WAVE;
  if (baseRow >= n_rows) return;  // wave-uniform

  if (baseRow + ROWS_PER_WAVE <= n_rows) {
    // ---- full-tile WMMA path (EXEC all ones throughout) ----
    // A-matrix 16x4 f32 layout (ISA 7.12.2):
    //   lanes 0-15  : VGPR0 = A[m][k0+0], VGPR1 = A[m][k0+1]   (m = lane)
    //   lanes 16-31 : VGPR0 = A[m][k0+2], VGPR1 = A[m][k0+3]   (m = lane-16)
    // B = s^T mirrors A with M<->N, so the identical per-lane b64 load
    // pattern feeds all three operands.
    const int rowInTile = lane & 15;
    const int colOff    = (lane >> 4) << 1;  // 0 or 2
    const long long elemOff = (baseRow + rowInTile) * (long long)D_DIM + colOff;

    const float* __restrict__ pim  = im     + elemOff;
    const float* __restrict__ pknn = knn_im + elemOff;
    const float* __restrict__ ps   = s      + elemOff;

    v8f acc_im  = {};
    v8f acc_knn = {};

    // Pipeline buffers: slot j holds the K-chunk at k = k0 + 4*j.
    v2f bufA[PF], bufK[PF], bufB[PF];
#pragma unroll
    for (int j = 0; j < PF; ++j) {
      bufA[j] = *(const v2f*)(pim  + 4 * j);
      bufK[j] = *(const v2f*)(pknn + 4 * j);
      bufB[j] = *(const v2f*)(ps   + 4 * j);
    }

    // Steady state: consume slot j, refill it PF steps ahead.
#pragma unroll 2
    for (int k0 = 0; k0 < D_DIM - 4 * PF; k0 += 4 * PF) {
#pragma unroll
      for (int j = 0; j < PF; ++j) {
        acc_im = __builtin_amdgcn_wmma_f32_16x16x4_f32(
            false, bufA[j], false, bufB[j], (short)0, acc_im, false, false);
        acc_knn = __builtin_amdgcn_wmma_f32_16x16x4_f32(
            false, bufK[j], false, bufB[j], (short)0, acc_knn, false, false);
        const int kn = k0 + 4 * PF + 4 * j;
        bufA[j] = *(const v2f*)(pim  + kn);
        bufK[j] = *(const v2f*)(pknn + kn);
        bufB[j] = *(const v2f*)(ps   + kn);
      }
    }
    // Drain: last PF slots, no reload.
#pragma unroll
    for (int j = 0; j < PF; ++j) {
      acc_im = __builtin_amdgcn_wmma_f32_16x16x4_f32(
          false, bufA[j], false, bufB[j], (short)0, acc_im, false, false);
      acc_knn = __builtin_amdgcn_wmma_f32_16x16x4_f32(
          false, bufK[j], false, bufB[j], (short)0, acc_knn, false, false);
    }

    // Diagonal extraction. C/D layout: VGPR r holds row M=r (lanes 0-15,
    // N=lane) and row M=8+r (lanes 16-31, N=lane-16). Diagonal (M==N):
    //   M=r   -> VGPR r, lane r
    //   M=8+r -> VGPR r, lane 24+r
    float d_im = 0.0f, d_knn = 0.0f;
#pragma unroll
    for (int r = 0; r < 8; ++r) {
      const bool sel = (lane == r) || (lane == 24 + r);
      if (sel) { d_im = acc_im[r]; d_knn = acc_knn[r]; }
    }
    const float cost = fmaxf(MARGIN_V + d_knn - d_im, 0.0f);
    if (lane < 8) {
      out[baseRow + lane] = cost;            // rows M=0..7
    } else if (lane >= 24) {
      out[baseRow + lane - 16] = cost;       // rows M=8..15
    }
  } else {
    // ---- ragged tail (never taken for N=262144): scalar per-lane rows ----
    const long long row = baseRow + lane;
    if (lane < ROWS_PER_WAVE && row < n_rows) {
      const float* __restrict__ pim  = im     + row * (long long)D_DIM;
      const float* __restrict__ pknn = knn_im + row * (long long)D_DIM;
      const float* __restrict__ ps   = s      + row * (long long)D_DIM;
      float d_im = 0.0f, d_knn = 0.0f;
      for (int k = 0; k < D_DIM; ++k) {
        const float sv = ps[k];
        d_im  += pim[k]  * sv;
        d_knn += pknn[k] * sv;
      }
      out[row] = fmaxf(MARGIN_V + d_knn - d_im, 0.0f);
    }
  }
}

extern "C" void kernel_launch(void* const* d_in, const int* in_sizes, int n_in,
                              void* d_out, int out_size, void* d_ws, size_t ws_size,
                              hipStream_t stream) {
  (void)n_in; (void)d_ws; (void)ws_size;
  const float* im     = (const float*)d_in[0];
  const float* knn_im = (const float*)d_in[1];
  const float* s      = (const float*)d_in[2];
  float* out = (float*)d_out;

  const int n_rows = in_sizes[0] / D_DIM;  // 262144
  (void)out_size;

  const int blocks = (n_rows + ROWS_PER_BLOCK - 1) / ROWS_PER_BLOCK;  // 2048
  knn_contrastive_wmma<<<blocks, ROWS_PER_BLOCK / ROWS_PER_WAVE * 32, 0, stream>>>(
      im, knn_im, s, out, n_rows);
}